// RightLeftMaxPooling_89593017794765
// MI455X (gfx1250) — compile-verified
//
#include <hip/hip_runtime.h>
#include <cstddef>

// Suffix max (reverse cummax) along rows of length 2048, f32.
// One wave32 per row; lane L owns the contiguous 64-float chunk [L*64, L*64+64).
// Memory-bound: ~256 MiB traffic -> ~11.5 us floor at 23.3 TB/s.
// Non-temporal hints on both loads and stores: dataset (256 MiB) > L2 (192 MiB),
// data is touched exactly once, so bypass/NT is the right cache policy.

typedef float v4f __attribute__((ext_vector_type(4)));

__global__ __launch_bounds__(256)
void suffix_max_rows_f32(const float* __restrict__ x,
                         float* __restrict__ y,
                         int rows) {
    constexpr int W     = 2048;     // row length
    constexpr int CHUNK = W / 32;   // 64 floats per lane
    constexpr int NV    = CHUNK / 4;// 16 float4 per lane

    const int lane = threadIdx.x & 31;
    const int row  = blockIdx.x * (blockDim.x >> 5) + (threadIdx.x >> 5);
    if (row >= rows) return;

    const v4f* __restrict__ src =
        reinterpret_cast<const v4f*>(x + (size_t)row * W) + (size_t)lane * NV;
    v4f* __restrict__ dst =
        reinterpret_cast<v4f*>(y + (size_t)row * W) + (size_t)lane * NV;

    // 1) Stream the lane's 8 KB-row share into registers (16 x global_load_b128, NT).
    v4f v[NV];
#pragma unroll
    for (int j = 0; j < NV; ++j) {
        v[j] = __builtin_nontemporal_load(src + j);
    }

    // 2) In-register reverse scan of the 64-element chunk (right -> left).
    float run = -__builtin_inff();
#pragma unroll
    for (int j = NV - 1; j >= 0; --j) {
        v4f t = v[j];
        t.w = __builtin_fmaxf(t.w, run);
        t.z = __builtin_fmaxf(t.z, t.w);
        t.y = __builtin_fmaxf(t.y, t.z);
        t.x = __builtin_fmaxf(t.x, t.y);
        v[j] = t;
        run  = t.x;          // chunk suffix-max so far
    }

    // 3) Wave32 inclusive suffix scan of chunk maxima (Hillis-Steele, 5 steps).
    //    After the loop: m[L] = max over lanes L..31 of chunk-max.
    float m = run;
#pragma unroll
    for (int off = 1; off < 32; off <<= 1) {
        float o = __shfl_down(m, off, 32);
        m = (lane + off < 32) ? __builtin_fmaxf(m, o) : m;
    }

    // 4) Exclusive-from-the-right value for this lane: e[L] = m[L+1], e[31] = -inf.
    float e = __shfl_down(m, 1, 32);
    if (lane == 31) e = -__builtin_inff();

    // 5) Fold in the cross-lane suffix max and stream out (NT stores).
#pragma unroll
    for (int j = 0; j < NV; ++j) {
        v4f t = v[j];
        t.x = __builtin_fmaxf(t.x, e);
        t.y = __builtin_fmaxf(t.y, e);
        t.z = __builtin_fmaxf(t.z, e);
        t.w = __builtin_fmaxf(t.w, e);
        __builtin_nontemporal_store(t, dst + j);
    }
}

extern "C" void kernel_launch(void* const* d_in, const int* in_sizes, int n_in,
                              void* d_out, int out_size, void* d_ws, size_t ws_size,
                              hipStream_t stream) {
    (void)n_in; (void)out_size; (void)d_ws; (void)ws_size;
    const float* x = (const float*)d_in[0];
    float*       y = (float*)d_out;

    const int W    = 2048;
    const int rows = in_sizes[0] / W;           // 8*1*2048 = 16384 rows

    const int waves_per_block = 8;              // 256 threads = 8 wave32
    dim3 block(256);
    dim3 grid((rows + waves_per_block - 1) / waves_per_block);

    hipLaunchKernelGGL(suffix_max_rows_f32, grid, block, 0, stream, x, y, rows);
}